// GAT_67001489817831
// MI455X (gfx1250) — compile-verified
//
#include <hip/hip_runtime.h>
#include <math.h>

#define NN   50000
#define NE   1600000
#define NEP  (NE + NN)      // edges + self loops = 1,650,000
#define FIN  128
#define HID  32
#define NH   3
#define C1   96             // HEADS*HID
#define C2   16             // OUT
#define SLOPE 0.2f

typedef float v2f __attribute__((ext_vector_type(2)));
typedef float v8f __attribute__((ext_vector_type(8)));

// Monotone float-max via native integer atomics (init memory to 0xFFFFFFFF = -NaN).
// Non-negative floats compare correctly as signed ints; negative floats compare
// inversely as unsigned. The two RMWs form a join-semilattice -> order independent.
__device__ __forceinline__ void atomic_max_f32(float* addr, float v) {
  if (v >= 0.0f) atomicMax((int*)addr, __float_as_int(v));
  else           atomicMin((unsigned int*)addr, (unsigned int)__float_as_int(v));
}

__device__ __forceinline__ void edge_sd(const int* __restrict__ ei, int eid, int& s, int& d) {
  if (eid < NE) { s = ei[eid]; d = ei[NE + eid]; }
  else          { s = eid - NE; d = s; }          // appended self-loops
}

// ---------------- GEMM1: H1[N,96] = X[N,128] @ W1[128,96] (fp32 WMMA) -------
// One wave per 16-row tile; 6 accumulators cover the 96-wide output.
// A 16x4 f32 fragment: lane(0-15)->K{k0,k0+1}, lane(16-31)->K{k0+2,k0+3}.
__global__ __launch_bounds__(256) void gemm1_wmma(const float* __restrict__ X,
                                                  const float* __restrict__ W,
                                                  float* __restrict__ H) {
  const int lane = threadIdx.x & 31;
  const int tile = blockIdx.x * 8 + (threadIdx.x >> 5);
  if (tile >= NN / 16) return;                    // wave-uniform, EXEC stays all-1
  const int arow = tile * 16 + (lane & 15);
  const int koff = (lane >> 4) << 1;
  const int col  = lane & 15;
  v8f acc[6];
  #pragma unroll
  for (int t = 0; t < 6; ++t) acc[t] = (v8f){};
  for (int k0 = 0; k0 < FIN; k0 += 4) {
    const int k = k0 + koff;
    v2f a; a.x = X[arow * FIN + k]; a.y = X[arow * FIN + k + 1];
    const float* Wk = W + k * C1;
    #pragma unroll
    for (int t = 0; t < 6; ++t) {
      v2f b; b.x = Wk[t * 16 + col]; b.y = Wk[C1 + t * 16 + col];
      acc[t] = __builtin_amdgcn_wmma_f32_16x16x4_f32(false, a, false, b,
                                                     (short)0, acc[t], false, false);
    }
  }
  const int m0 = (lane >> 4) << 3;                // C/D: lanes 16-31 hold rows 8..15
  #pragma unroll
  for (int i = 0; i < 8; ++i) {
    float* Hr = H + (size_t)(tile * 16 + m0 + i) * C1;
    #pragma unroll
    for (int t = 0; t < 6; ++t) Hr[t * 16 + col] = acc[t][i];
  }
}

// ---------------- GEMM2: H2[N,16] = H1relu[N,96] @ W2[96,16] (fp32 WMMA) ----
__global__ __launch_bounds__(256) void gemm2_wmma(const float* __restrict__ Hin,
                                                  const float* __restrict__ W,
                                                  float* __restrict__ Hout) {
  const int lane = threadIdx.x & 31;
  const int tile = blockIdx.x * 8 + (threadIdx.x >> 5);
  if (tile >= NN / 16) return;
  const int arow = tile * 16 + (lane & 15);
  const int koff = (lane >> 4) << 1;
  const int col  = lane & 15;
  v8f acc = (v8f){};
  for (int k0 = 0; k0 < C1; k0 += 4) {
    const int k = k0 + koff;
    v2f a; a.x = Hin[arow * C1 + k]; a.y = Hin[arow * C1 + k + 1];
    v2f b; b.x = W[k * C2 + col];    b.y = W[(k + 1) * C2 + col];
    acc = __builtin_amdgcn_wmma_f32_16x16x4_f32(false, a, false, b,
                                                (short)0, acc, false, false);
  }
  const int m0 = (lane >> 4) << 3;
  #pragma unroll
  for (int i = 0; i < 8; ++i)
    Hout[(size_t)(tile * 16 + m0 + i) * C2 + col] = acc[i];
}

// ---------------- attention logits ------------------------------------------
__global__ void attn1_kernel(const float* __restrict__ H, const float* __restrict__ as,
                             const float* __restrict__ ad, float* __restrict__ asrc,
                             float* __restrict__ adst) {
  int idx = blockIdx.x * blockDim.x + threadIdx.x;
  if (idx >= NN * NH) return;
  int n = idx / NH, h = idx - n * NH;
  const float* hp = H + (size_t)n * C1 + h * HID;
  const float* sp = as + h * HID;
  const float* dp = ad + h * HID;
  float s = 0.f, d = 0.f;
  #pragma unroll
  for (int f = 0; f < HID; ++f) { float v = hp[f]; s += v * sp[f]; d += v * dp[f]; }
  asrc[idx] = s; adst[idx] = d;
}

__global__ void attn2_kernel(const float* __restrict__ H, const float* __restrict__ as,
                             const float* __restrict__ ad, float* __restrict__ asrc,
                             float* __restrict__ adst) {
  int n = blockIdx.x * blockDim.x + threadIdx.x;
  if (n >= NN) return;
  const float* hp = H + (size_t)n * C2;
  float s = 0.f, d = 0.f;
  #pragma unroll
  for (int f = 0; f < C2; ++f) { float v = hp[f]; s += v * as[f]; d += v * ad[f]; }
  asrc[n] = s; adst[n] = d;
}

// ---------------- layer-1 edge passes ---------------------------------------
__global__ void edge_max1(const int* __restrict__ ei, const float* __restrict__ asrc,
                          const float* __restrict__ adst, float* __restrict__ e1,
                          float* __restrict__ mx) {
  int idx = blockIdx.x * blockDim.x + threadIdx.x;
  if (idx >= NEP * NH) return;
  int eid = idx / NH, h = idx - eid * NH;
  int s, d; edge_sd(ei, eid, s, d);
  float e = asrc[s * NH + h] + adst[d * NH + h];
  e = e > 0.f ? e : SLOPE * e;
  e1[idx] = e;
  atomic_max_f32(&mx[d * NH + h], e);
}

__global__ void edge_exp1(const int* __restrict__ ei, float* __restrict__ e1,
                          const float* __restrict__ mx, float* __restrict__ den) {
  int idx = blockIdx.x * blockDim.x + threadIdx.x;
  if (idx >= NEP * NH) return;
  int eid = idx / NH, h = idx - eid * NH;
  int s, d; edge_sd(ei, eid, s, d);
  float e = __expf(e1[idx] - mx[d * NH + h]);
  e1[idx] = e;
  atomicAdd(&den[d * NH + h], e);
}

__global__ void edge_agg1(const int* __restrict__ ei, const float* __restrict__ e1,
                          const float* __restrict__ den, const float* __restrict__ H,
                          float* __restrict__ Out) {
  int idx = blockIdx.x * blockDim.x + threadIdx.x;
  if (idx >= NEP * C1) return;
  int eid = idx / C1, c = idx - eid * C1, h = c >> 5;
  int s, d; edge_sd(ei, eid, s, d);
  float alpha = e1[eid * NH + h] / den[d * NH + h];
  atomicAdd(&Out[(size_t)d * C1 + c], alpha * H[(size_t)s * C1 + c]);
}

__global__ void relu_bias1(float* __restrict__ Out, const float* __restrict__ bias) {
  int idx = blockIdx.x * blockDim.x + threadIdx.x;
  if (idx >= NN * C1) return;
  int c = idx % C1;
  float v = Out[idx] + bias[c];
  Out[idx] = v > 0.f ? v : 0.f;
}

// ---------------- layer-2 edge passes (single head) -------------------------
__global__ void edge_max2(const int* __restrict__ ei, const float* __restrict__ asrc,
                          const float* __restrict__ adst, float* __restrict__ e2,
                          float* __restrict__ mx) {
  int eid = blockIdx.x * blockDim.x + threadIdx.x;
  if (eid >= NEP) return;
  int s, d; edge_sd(ei, eid, s, d);
  float e = asrc[s] + adst[d];
  e = e > 0.f ? e : SLOPE * e;
  e2[eid] = e;
  atomic_max_f32(&mx[d], e);
}

__global__ void edge_exp2(const int* __restrict__ ei, float* __restrict__ e2,
                          const float* __restrict__ mx, float* __restrict__ den) {
  int eid = blockIdx.x * blockDim.x + threadIdx.x;
  if (eid >= NEP) return;
  int s, d; edge_sd(ei, eid, s, d);
  float e = __expf(e2[eid] - mx[d]);
  e2[eid] = e;
  atomicAdd(&den[d], e);
}

__global__ void edge_agg2(const int* __restrict__ ei, const float* __restrict__ e2,
                          const float* __restrict__ den, const float* __restrict__ H,
                          float* __restrict__ Out) {
  int idx = blockIdx.x * blockDim.x + threadIdx.x;
  if (idx >= NEP * C2) return;
  int eid = idx >> 4, c = idx & 15;
  int s, d; edge_sd(ei, eid, s, d);
  float alpha = e2[eid] / den[d];
  atomicAdd(&Out[(size_t)d * C2 + c], alpha * H[(size_t)s * C2 + c]);
}

// ---------------- final bias + log_softmax ----------------------------------
__global__ void logsoftmax_kernel(const float* __restrict__ Out2,
                                  const float* __restrict__ bias,
                                  float* __restrict__ y) {
  int n = blockIdx.x * blockDim.x + threadIdx.x;
  if (n >= NN) return;
  float v[C2];
  float m = -3.0e38f;
  #pragma unroll
  for (int c = 0; c < C2; ++c) { v[c] = Out2[(size_t)n * C2 + c] + bias[c]; m = fmaxf(m, v[c]); }
  float ssum = 0.f;
  #pragma unroll
  for (int c = 0; c < C2; ++c) ssum += __expf(v[c] - m);
  float ls = __logf(ssum);
  #pragma unroll
  for (int c = 0; c < C2; ++c) y[(size_t)n * C2 + c] = v[c] - m - ls;
}

extern "C" void kernel_launch(void* const* d_in, const int* in_sizes, int n_in,
                              void* d_out, int out_size, void* d_ws, size_t ws_size,
                              hipStream_t stream) {
  const float* x   = (const float*)d_in[0];
  const int*   ei  = (const int*)d_in[1];
  const float* W1  = (const float*)d_in[2];
  const float* as1 = (const float*)d_in[3];
  const float* ad1 = (const float*)d_in[4];
  const float* b1  = (const float*)d_in[5];
  const float* W2  = (const float*)d_in[6];
  const float* as2 = (const float*)d_in[7];
  const float* ad2 = (const float*)d_in[8];
  const float* b2  = (const float*)d_in[9];
  float* y = (float*)d_out;

  float* ws = (float*)d_ws;
  size_t off = 0;
  float* h1    = ws + off; off += (size_t)NN * C1;   // 4.80M
  float* h1o   = ws + off; off += (size_t)NN * C1;   // 4.80M
  float* asrc1 = ws + off; off += (size_t)NN * NH;
  float* adst1 = ws + off; off += (size_t)NN * NH;
  float* mx1   = ws + off; off += (size_t)NN * NH;
  float* den1  = ws + off; off += (size_t)NN * NH;
  float* e1    = ws + off; off += (size_t)NEP * NH;  // 4.95M
  float* h2    = ws + off; off += (size_t)NN * C2;
  float* asrc2 = ws + off; off += (size_t)NN;
  float* adst2 = ws + off; off += (size_t)NN;
  float* mx2   = ws + off; off += (size_t)NN;
  float* den2  = ws + off; off += (size_t)NN;
  float* e2    = ws + off; off += (size_t)NEP;       // 1.65M
  float* out2  = ws + off; off += (size_t)NN * C2;   // total ~18.6M floats ~75MB

  hipMemsetAsync(h1o,  0,    sizeof(float) * (size_t)NN * C1, stream);
  hipMemsetAsync(den1, 0,    sizeof(float) * (size_t)NN * NH, stream);
  hipMemsetAsync(mx1,  0xFF, sizeof(float) * (size_t)NN * NH, stream);  // -NaN ~ -inf seed
  hipMemsetAsync(out2, 0,    sizeof(float) * (size_t)NN * C2, stream);
  hipMemsetAsync(den2, 0,    sizeof(float) * (size_t)NN, stream);
  hipMemsetAsync(mx2,  0xFF, sizeof(float) * (size_t)NN, stream);

  const int B = 256;
  gemm1_wmma<<<(NN / 16 + 7) / 8, B, 0, stream>>>(x, W1, h1);
  attn1_kernel<<<(NN * NH + B - 1) / B, B, 0, stream>>>(h1, as1, ad1, asrc1, adst1);
  edge_max1<<<(int)(((size_t)NEP * NH + B - 1) / B), B, 0, stream>>>(ei, asrc1, adst1, e1, mx1);
  edge_exp1<<<(int)(((size_t)NEP * NH + B - 1) / B), B, 0, stream>>>(ei, e1, mx1, den1);
  edge_agg1<<<(int)(((size_t)NEP * C1 + B - 1) / B), B, 0, stream>>>(ei, e1, den1, h1, h1o);
  relu_bias1<<<(NN * C1 + B - 1) / B, B, 0, stream>>>(h1o, b1);
  gemm2_wmma<<<(NN / 16 + 7) / 8, B, 0, stream>>>(h1o, W2, h2);
  attn2_kernel<<<(NN + B - 1) / B, B, 0, stream>>>(h2, as2, ad2, asrc2, adst2);
  edge_max2<<<(NEP + B - 1) / B, B, 0, stream>>>(ei, asrc2, adst2, e2, mx2);
  edge_exp2<<<(NEP + B - 1) / B, B, 0, stream>>>(ei, e2, mx2, den2);
  edge_agg2<<<(int)(((size_t)NEP * C2 + B - 1) / B), B, 0, stream>>>(ei, e2, den2, h2, out2);
  logsoftmax_kernel<<<(NN + B - 1) / B, B, 0, stream>>>(out2, b2, y);
}